// SpGraphAttentionLayer_90168543412402
// MI455X (gfx1250) — compile-verified
//
#include <hip/hip_runtime.h>
#include <cfloat>

#define N_NODES 100000
#define N_EDGES 3200000
#define DIM     128
#define KC      10
#define NITER   10
#define XPAD    132   // LDS row stride in floats (bank-conflict-free, 16B aligned)

typedef __attribute__((ext_vector_type(2))) float v2f;
typedef __attribute__((ext_vector_type(8))) float v8f;

// ---- workspace layout (in floats) ----
#define OFF_C      0        // 16*128 centroids (rows 10..15 zero-padded)
#define OFF_SUMS   2048     // 16*128
#define OFF_COUNTS 4096     // 16 (contiguous after sums)
#define OFF_TABLE  4112     // 16*16 centroid attention table
#define OFF_CL     4368     // 100000 int32 cluster assignments
#define OFF_SEGMAX 104368   // 100000
#define OFF_SEGSUM 204368   // 100000 (contiguous after segmax)

__global__ void zero_f32(float* __restrict__ p, int n) {
    int i = blockIdx.x * blockDim.x + threadIdx.x;
    int st = gridDim.x * blockDim.x;
    for (; i < n; i += st) p[i] = 0.0f;
}

__global__ void init_centroids(const float* __restrict__ x, float* __restrict__ c) {
    for (int i = threadIdx.x; i < 16 * DIM; i += blockDim.x) {
        int row = i >> 7;
        c[i] = (row < KC) ? x[i] : 0.0f;   // c = x[:10]; rows 10..15 = 0
    }
}

// One wave per 16-row tile of x; 4 waves (= 64 rows) per block.
// dist[m][n] = -2 * dot(x_m, c_n) + ||c_n||^2, argmin over n in [0,10).
__global__ __launch_bounds__(128)
void assign_clusters(const float* __restrict__ x, const float* __restrict__ c,
                     int* __restrict__ cl) {
    __shared__ __align__(16) float cB[DIM * 16];      // B[k][n] = c[n][k]
    __shared__ float cnorm[16];
    __shared__ __align__(16) float xs[4 * 16 * XPAD]; // per-wave x tiles

    const int tid  = threadIdx.x;
    const int lane = tid & 31;
    const int wave = tid >> 5;
    const int tileBase = (blockIdx.x * 4 + wave) * 16;
    float* xsw = &xs[wave * 16 * XPAD];

    // phase 1: stage transposed centroids + own x tile, zero cnorm
    for (int i = tid; i < DIM * 16; i += 128) {
        int k = i >> 4, n = i & 15;
        cB[i] = c[n * DIM + k];
    }
    if (tid < 16) cnorm[tid] = 0.0f;
    for (int r = 0; r < 16; ++r) {
        int row = tileBase + r;
        row = row < N_NODES ? row : (N_NODES - 1);     // clamp; writes masked later
        const float4 v = ((const float4*)(x + (size_t)row * DIM))[lane];
        *((float4*)&xsw[r * XPAD + lane * 4]) = v;     // coalesced 512B per row
    }
    __syncthreads();

    // phase 2: ||c_n||^2 via LDS atomics
    {
        int n = tid & 15, ks = tid >> 4;
        float p = 0.0f;
        for (int k = ks; k < DIM; k += 8) { float t = cB[k * 16 + n]; p += t * t; }
        atomicAdd(&cnorm[n], p);
    }
    __syncthreads();

    // phase 3: 32 x V_WMMA_F32_16X16X4_F32 over K=128
    const int mn    = lane & 15;            // A row m / B col n for this lane
    const int khalf = (lane >> 4) << 1;     // lanes 16..31 hold K+2,K+3
    v8f acc = {0.f, 0.f, 0.f, 0.f, 0.f, 0.f, 0.f, 0.f};
#pragma unroll
    for (int k0 = 0; k0 < DIM; k0 += 4) {
        const int kk = k0 + khalf;
        float2 af = *((const float2*)&xsw[mn * XPAD + kk]);     // ds_load_b64, conflict-free
        v2f a; a.x = af.x; a.y = af.y;                          // A[m][kk], A[m][kk+1]
        v2f b; b.x = cB[kk * 16 + mn]; b.y = cB[(kk + 1) * 16 + mn]; // B[kk][n], B[kk+1][n]
        acc = __builtin_amdgcn_wmma_f32_16x16x4_f32(
                  false, a, false, b, (short)0, acc, false, false);
    }

    // argmin over n per row; C layout: lane = N, VGPR r -> M = r + 8*(lane>>4)
    const float cn = cnorm[mn];
#pragma unroll
    for (int r = 0; r < 8; ++r) {
        float val = (mn < KC) ? fmaf(-2.0f, acc[r], cn) : FLT_MAX;
        int   idx = mn;
#pragma unroll
        for (int s = 1; s < 16; s <<= 1) {       // butterfly within each 16-lane half
            float ov = __shfl_xor(val, s, 32);
            int   oi = __shfl_xor(idx, s, 32);
            if (ov < val || (ov == val && oi < idx)) { val = ov; idx = oi; }
        }
        if (mn == 0) {
            int row = tileBase + r + ((lane >> 4) << 3);
            if (row < N_NODES) cl[row] = idx;
        }
    }
}

// Block-local LDS accumulation of per-cluster sums/counts, one global flush.
__global__ __launch_bounds__(256)
void accumulate(const float* __restrict__ x, const int* __restrict__ cl,
                float* __restrict__ sums, float* __restrict__ counts) {
    __shared__ float ls[KC * DIM];
    __shared__ float lc[KC];
    for (int i = threadIdx.x; i < KC * DIM; i += 256) ls[i] = 0.0f;
    if (threadIdx.x < KC) lc[threadIdx.x] = 0.0f;
    __syncthreads();
    const int d = threadIdx.x & 127;
    const int r = threadIdx.x >> 7;    // 2 node-rows per block
    for (int node = blockIdx.x * 2 + r; node < N_NODES; node += gridDim.x * 2) {
        int j = cl[node];
        atomicAdd(&ls[j * DIM + d], x[(size_t)node * DIM + d]);  // ds_add_f32
        if (d == 0) atomicAdd(&lc[j], 1.0f);
    }
    __syncthreads();
    for (int i = threadIdx.x; i < KC * DIM; i += 256) atomicAdd(&sums[i], ls[i]);
    if (threadIdx.x < KC) atomicAdd(&counts[threadIdx.x], lc[threadIdx.x]);
}

__global__ void update_centroids(const float* __restrict__ sums,
                                 const float* __restrict__ counts,
                                 float* __restrict__ c) {
    for (int i = threadIdx.x; i < 16 * DIM; i += blockDim.x) {
        int j = i >> 7;
        c[i] = sums[i] / fmaxf(counts[j], 1.0f);   // padded rows stay 0
    }
}

// q = c@WQ, v = c@WV, prods = q@v.T/sqrt(128), row softmax -> table[16][16]
__global__ __launch_bounds__(256)
void attention_table(const float* __restrict__ c, const float* __restrict__ WQ,
                     const float* __restrict__ WV, float* __restrict__ table) {
    __shared__ float qs[KC * DIM];
    __shared__ float vs[KC * DIM];
    __shared__ float ps[KC * 16];
    for (int o = threadIdx.x; o < KC * DIM; o += 256) {
        int j = o >> 7, d = o & 127;
        float q = 0.0f, v = 0.0f;
        for (int k = 0; k < DIM; ++k) {
            float ck = c[j * DIM + k];
            q = fmaf(ck, WQ[k * DIM + d], q);
            v = fmaf(ck, WV[k * DIM + d], v);
        }
        qs[o] = q; vs[o] = v;
    }
    __syncthreads();
    const float scale = rsqrtf((float)DIM);
    for (int o = threadIdx.x; o < KC * KC; o += 256) {
        int i = o / KC, j = o % KC;
        float p = 0.0f;
        for (int d = 0; d < DIM; ++d) p = fmaf(qs[i * DIM + d], vs[j * DIM + d], p);
        ps[i * 16 + j] = p * scale;
    }
    __syncthreads();
    if (threadIdx.x < KC) {
        int i = threadIdx.x;
        float m = -FLT_MAX;
        for (int j = 0; j < KC; ++j) m = fmaxf(m, ps[i * 16 + j]);
        float e[KC]; float s = 0.0f;
        for (int j = 0; j < KC; ++j) { e[j] = expf(ps[i * 16 + j] - m); s += e[j]; }
        for (int j = 0; j < KC; ++j) table[i * 16 + j] = e[j] / s;
    }
}

// pass 1: gather table value per edge, segment max via uint atomicMax (vals > 0)
__global__ __launch_bounds__(256)
void edge_pass1(const long long* __restrict__ edge, const int* __restrict__ cl,
                const float* __restrict__ table, float* __restrict__ out,
                unsigned* __restrict__ segmax) {
    int e = blockIdx.x * 256 + threadIdx.x;
    if (e >= N_EDGES) return;
    int s = (int)edge[e];
    int d = (int)edge[N_EDGES + e];
    float val = table[cl[s] * 16 + cl[d]];
    out[e] = val;
    atomicMax(&segmax[s], __float_as_uint(val));
}

__global__ __launch_bounds__(256)
void edge_pass2(const long long* __restrict__ edge, const float* __restrict__ segmax,
                float* __restrict__ out, float* __restrict__ segsum) {
    int e = blockIdx.x * 256 + threadIdx.x;
    if (e >= N_EDGES) return;
    int s = (int)edge[e];
    float ex = expf(out[e] - segmax[s]);
    out[e] = ex;
    atomicAdd(&segsum[s], ex);
}

__global__ __launch_bounds__(256)
void edge_pass3(const long long* __restrict__ edge, const float* __restrict__ segsum,
                float* __restrict__ out) {
    int e = blockIdx.x * 256 + threadIdx.x;
    if (e >= N_EDGES) return;
    int s = (int)edge[e];
    out[e] = out[e] / (segsum[s] + 1e-16f);
}

extern "C" void kernel_launch(void* const* d_in, const int* in_sizes, int n_in,
                              void* d_out, int out_size, void* d_ws, size_t ws_size,
                              hipStream_t stream) {
    const float*     x    = (const float*)d_in[0];
    const long long* edge = (const long long*)d_in[1];   // int64, shape (2, N_EDGES)
    const float*     WQ   = (const float*)d_in[2];
    const float*     WV   = (const float*)d_in[3];
    // d_in[4] = WK: only its shape (128) is used -> sqrt(128) constant

    float* ws     = (float*)d_ws;
    float* c      = ws + OFF_C;
    float* sums   = ws + OFF_SUMS;
    float* counts = ws + OFF_COUNTS;
    float* table  = ws + OFF_TABLE;
    int*   cl     = (int*)(ws + OFF_CL);
    float* segmax = ws + OFF_SEGMAX;
    float* segsum = ws + OFF_SEGSUM;
    float* out    = (float*)d_out;

    const int assignBlocks = (N_NODES + 63) / 64;       // 1563, 64 rows/block
    const int edgeBlocks   = (N_EDGES + 255) / 256;     // 12500

    init_centroids<<<1, 256, 0, stream>>>(x, c);
    for (int it = 0; it < NITER; ++it) {
        zero_f32<<<9, 256, 0, stream>>>(sums, 16 * DIM + 16);        // sums + counts
        assign_clusters<<<assignBlocks, 128, 0, stream>>>(x, c, cl);
        accumulate<<<512, 256, 0, stream>>>(x, cl, sums, counts);
        update_centroids<<<1, 256, 0, stream>>>(sums, counts, c);
    }
    attention_table<<<1, 256, 0, stream>>>(c, WQ, WV, table);
    zero_f32<<<782, 256, 0, stream>>>(segmax, 2 * N_NODES);          // segmax + segsum
    edge_pass1<<<edgeBlocks, 256, 0, stream>>>(edge, cl, table, out, (unsigned*)segmax);
    edge_pass2<<<edgeBlocks, 256, 0, stream>>>(edge, segmax, out, segsum);
    edge_pass3<<<edgeBlocks, 256, 0, stream>>>(edge, segsum, out);
}